// unit_gcn_22325240005244
// MI455X (gfx1250) — compile-verified
//
#include <hip/hip_runtime.h>

// ---------------------------------------------------------------------------
// Fused unit-GCN: y = relu(BN(sum_s (x @ (A_base+PA)[s]) conv1x1 Wd[s] + bias) + x)
// N=64, C=64, T=300, V=25, S=3.  fp32 end-to-end via V_WMMA_F32_16X16X4_F32.
// Round 3: W2 staging via global_load_async_to_lds_b128 (16B-aligned chunks
// on both global and LDS sides); single asynccnt drain for all staging.
// ---------------------------------------------------------------------------

typedef float v2f __attribute__((ext_vector_type(2)));
typedef float v8f __attribute__((ext_vector_type(8)));

#define GN 64     // batch
#define GC 64     // channels
#define GT 300    // time
#define GV 25     // vertices
#define TTILE 16  // t values per block
#define KP 28     // stage-1 K (V'=25) padded to multiple of 4
#define NP 32     // stage-1 N (W=25) padded to 2x16
#define W2P 196   // row stride of fused weight matrix (192 + 4 pad)
#define AGP 18    // row stride of per-wave agg staging buffer (16 + 2 pad)

#define XS_N (TTILE * GC * KP)      // 28672 floats
#define W2_N (GC * W2P)             // 12544 floats
#define AT_N (3 * NP * KP)          // 2688 floats
#define AGG_N (8 * 2 * NP * AGP)    // 9216 floats (ping-pong per wave)
#define SMEM_FLOATS (XS_N + W2_N + AT_N + AGG_N + 128)

static __device__ __forceinline__ v8f wmma4(v2f a, v2f b, v8f c) {
  // D(16x16,f32) = A(16x4,f32) x B(4x16,f32) + C
  return __builtin_amdgcn_wmma_f32_16x16x4_f32(false, a, false, b, (short)0, c,
                                               false, false);
}

extern "C" __global__ __launch_bounds__(256) void gcn_fused_kernel(
    const float* __restrict__ x, const float* __restrict__ Ab,
    const float* __restrict__ PA, const float* __restrict__ Wd,
    const float* __restrict__ bd, const float* __restrict__ gamma,
    const float* __restrict__ beta, const float* __restrict__ rmean,
    const float* __restrict__ rvar, float* __restrict__ out) {
  extern __shared__ float smem[];
  float* xs = smem;         // [TTILE][GC][KP]   x tile, zero padded
  float* W2 = xs + XS_N;    // [GC][W2P]         W2[o][s*64+c] = Wd[s][o][c]
  float* At = W2 + W2_N;    // [3][NP][KP]       At[s][w][v'] = A_full[s][v'][w]
  float* agg = At + AT_N;   // [8][2][NP][AGP]   per-wave ping-pong staging
  float* scof = agg + AGG_N;  // [64][2] (scale, offset) pairs

  const int tid = threadIdx.x;
  const int lane = tid & 31;
  const int wave = tid >> 5;
  const int lhalf = lane >> 4;  // 0 | 1 (selects K pair / M+8 half)
  const int l16 = lane & 15;
  const int n = blockIdx.y;
  const int t0 = blockIdx.x * TTILE;

  // ---- cooperative staging --------------------------------------------------
  // x tile via async global->LDS b32 (GVS mode: SGPR base + 32-bit VGPR off)
  {
    const float* xbase = x + (size_t)n * GC * GT * GV;
    for (int i = tid; i < XS_N; i += 256) {
      int tt = i / (GC * KP);
      int r = i % (GC * KP);
      int c = r / KP;
      int k = r % KP;
      int t = t0 + tt;
      if (k < GV && t < GT) {
        unsigned ldsoff = (unsigned)(size_t)(xs + i);
        unsigned voff = (unsigned)(((c * GT + t) * GV + k) * 4);
        asm volatile("global_load_async_to_lds_b32 %0, %1, %2"
                     :
                     : "v"(ldsoff), "v"(voff), "s"(xbase)
                     : "memory");
      } else {
        xs[i] = 0.0f;
      }
    }
  }
  // W2 gather via async b128: both sides 16B aligned
  //   global: Wd + ((s*64+o)*64 + q*4)*4  = (s*64+o)*256 + q*16
  //   LDS   : W2 + o*W2P + s*64 + q*4     -> byte o*784 + s*256 + q*16
  {
    for (int i = tid; i < GC * 3 * 16; i += 256) {
      int o = i / 48;
      int r = i % 48;
      int s = r / 16;
      int q = r % 16;  // 16-byte chunk within a 64-float Wd row
      unsigned ldsoff = (unsigned)(size_t)(W2 + o * W2P + s * 64 + q * 4);
      unsigned voff = (unsigned)((((s * GC + o) * GC) + q * 4) * 4);
      asm volatile("global_load_async_to_lds_b128 %0, %1, %2"
                   :
                   : "v"(ldsoff), "v"(voff), "s"(Wd)
                   : "memory");
    }
  }
  // A^T (needs arithmetic: A_base + PA, pad) and folded BN constants: normal path
  for (int i = tid; i < AT_N; i += 256) {
    int s = i / (NP * KP);
    int r = i % (NP * KP);
    int w = r / KP;
    int k = r % KP;
    float v = 0.0f;
    if (w < GV && k < GV) {
      int gi = (s * GV + k) * GV + w;
      v = Ab[gi] + PA[gi];
    }
    At[i] = v;
  }
  if (tid < GC) {
    float sc = gamma[tid] * rsqrtf(rvar[tid] + 1e-5f);
    float bsum = bd[tid] + bd[GC + tid] + bd[2 * GC + tid];
    scof[2 * tid] = sc;
    scof[2 * tid + 1] = (bsum - rmean[tid]) * sc + beta[tid];
  }
  asm volatile("s_wait_asynccnt 0" ::: "memory");
  __syncthreads();

  // ---- register-resident stage-1 B fragments (invariant over t and ct) ------
  v2f atf[3][2][7];
#pragma unroll
  for (int s = 0; s < 3; ++s)
#pragma unroll
    for (int nt = 0; nt < 2; ++nt)
#pragma unroll
      for (int kk = 0; kk < 7; ++kk)
        atf[s][nt][kk] =
            *(const v2f*)&At[(s * NP + nt * 16 + l16) * KP + kk * 4 + lhalf * 2];

  float* aggw = agg + wave * 2 * NP * AGP;

  for (int tpass = 0; tpass < 2; ++tpass) {
    const int tt = wave + tpass * 8;
    const int t = t0 + tt;
    const float* xrow = xs + tt * GC * KP;

    v8f acc[4][2];
#pragma unroll
    for (int ot = 0; ot < 4; ++ot) {
      acc[ot][0] = {};
      acc[ot][1] = {};
    }

    for (int ct = 0; ct < 4; ++ct) {
      // x A-fragments for this c-tile: reused across all 3 subsets x 2 n-tiles
      v2f xa[7];
#pragma unroll
      for (int kk = 0; kk < 7; ++kk)
        xa[kk] = *(const v2f*)&xrow[(ct * 16 + l16) * KP + kk * 4 + lhalf * 2];

#pragma unroll
      for (int s = 0; s < 3; ++s) {
        float* buf = aggw + ((ct * 3 + s) & 1) * NP * AGP;  // ping-pong
        // ---- stage 1: agg[16c][32w] = x[16c][25v'] @ A_s[25v'][32w] ---------
#pragma unroll
        for (int nt = 0; nt < 2; ++nt) {
          v8f d = {};
#pragma unroll
          for (int kk = 0; kk < 7; ++kk) d = wmma4(xa[kk], atf[s][nt][kk], d);
          // D layout: vgpr r -> (M = r + 8*lhalf, Ncol = l16); stage as [w][c]
#pragma unroll
          for (int r = 0; r < 8; ++r)
            buf[(nt * 16 + l16) * AGP + (r + lhalf * 8)] = d[r];
        }
        // ---- stage 2: acc[64o][32w] += W2[64o][16k] @ agg[16k][32w] ---------
        const int kbase = s * 64 + ct * 16;
#pragma unroll
        for (int kk4 = 0; kk4 < 4; ++kk4) {
          const int kl = kk4 * 4 + lhalf * 2;
          v2f b0 = *(const v2f*)&buf[l16 * AGP + kl];
          v2f b1 = *(const v2f*)&buf[(16 + l16) * AGP + kl];
#pragma unroll
          for (int ot = 0; ot < 4; ++ot) {
            v2f a = *(const v2f*)&W2[(ot * 16 + l16) * W2P + kbase + kl];
            acc[ot][0] = wmma4(a, b0, acc[ot][0]);
            acc[ot][1] = wmma4(a, b1, acc[ot][1]);
          }
        }
      }
    }

    // ---- epilogue: BN + bias + residual + ReLU ------------------------------
    if (t < GT) {
#pragma unroll
      for (int ot = 0; ot < 4; ++ot) {
#pragma unroll
        for (int r = 0; r < 8; ++r) {
          const int o = ot * 16 + r + lhalf * 8;
          v2f so = *(const v2f*)&scof[o * 2];  // (scale, offset)
#pragma unroll
          for (int nt = 0; nt < 2; ++nt) {
            const int v = nt * 16 + l16;
            if (v < GV) {
              float y = acc[ot][nt][r] * so[0] + so[1] + xrow[o * KP + v];
              y = fmaxf(y, 0.0f);
              out[((n * GC + o) * GT + t) * GV + v] = y;
            }
          }
        }
      }
    }
  }
}

// ---------------------------------------------------------------------------
extern "C" void kernel_launch(void* const* d_in, const int* in_sizes, int n_in,
                              void* d_out, int out_size, void* d_ws,
                              size_t ws_size, hipStream_t stream) {
  (void)in_sizes;
  (void)n_in;
  (void)out_size;
  (void)d_ws;
  (void)ws_size;

  const float* x = (const float*)d_in[0];
  const float* Ab = (const float*)d_in[1];
  const float* PA = (const float*)d_in[2];
  const float* Wd = (const float*)d_in[3];
  const float* bd = (const float*)d_in[4];
  const float* gamma = (const float*)d_in[5];
  const float* beta = (const float*)d_in[6];
  const float* rmean = (const float*)d_in[7];
  const float* rvar = (const float*)d_in[8];
  float* out = (float*)d_out;

  dim3 grid((GT + TTILE - 1) / TTILE, GN);  // 19 x 64
  dim3 block(256);
  size_t shmem = (size_t)SMEM_FLOATS * sizeof(float);  // ~208 KB of 320 KB WGP LDS
  hipLaunchKernelGGL(gcn_fused_kernel, grid, block, shmem, stream, x, Ab, PA,
                     Wd, bd, gamma, beta, rmean, rvar, out);
}